// SGC_48129403519234
// MI455X (gfx1250) — compile-verified
//
#include <hip/hip_runtime.h>
#include <hip/hip_bf16.h>

typedef __attribute__((ext_vector_type(2))) float v2f;
typedef __attribute__((ext_vector_type(8))) float v8f;

#define IN_CH 128
#define OUT_CH 128

static __device__ __forceinline__ v8f wmma4_f32(v2f a, v2f b, v8f c) {
    // D = A(16x4) x B(4x16) + C(16x16), all fp32
    return __builtin_amdgcn_wmma_f32_16x16x4_f32(
        /*neg_a=*/false, a, /*neg_b=*/false, b,
        /*c_mod=*/(short)0, c, /*reuse_a=*/false, /*reuse_b=*/false);
}

// ---------------- degree / normalization ----------------

__global__ void deg_init_kernel(float* deg, int n) {
    int i = blockIdx.x * blockDim.x + threadIdx.x;
    if (i < n) deg[i] = 1.0f;   // self-loop contributes 1 to every node
}

__global__ void deg_accum_kernel(const int* __restrict__ dst, float* deg, int e) {
    int i = blockIdx.x * blockDim.x + threadIdx.x;
    if (i < e) atomicAdd(&deg[dst[i]], 1.0f);
}

__global__ void deg_rsqrt_kernel(float* deg, int n) {
    int i = blockIdx.x * blockDim.x + threadIdx.x;
    if (i < n) deg[i] = rsqrtf(deg[i]);   // deg >= 1 always (self-loops)
}

// ---------------- propagation: out = A_hat * h ----------------
// init handles the self-loop edge: out[i] = h[i] * dinv[i]^2

__global__ void hop_init_kernel(const float* __restrict__ h,
                                const float* __restrict__ dinv,
                                float* __restrict__ out, int n) {
    int tid = blockIdx.x * blockDim.x + threadIdx.x;
    int i = tid >> 5;          // node
    int lane = tid & 31;       // 32 lanes x float4 = 128 channels
    if (i >= n) return;
    float wi = dinv[i] * dinv[i];
    float4 v = ((const float4*)(h + (size_t)i * IN_CH))[lane];
    float4 o; o.x = v.x * wi; o.y = v.y * wi; o.z = v.z * wi; o.w = v.w * wi;
    ((float4*)(out + (size_t)i * IN_CH))[lane] = o;
}

__global__ void hop_scatter_kernel(const float* __restrict__ h,
                                   const float* __restrict__ dinv,
                                   const int* __restrict__ src,
                                   const int* __restrict__ dst,
                                   float* __restrict__ out, int e) {
    int tid = blockIdx.x * blockDim.x + threadIdx.x;
    int ed = tid >> 5;
    int lane = tid & 31;
    if (ed >= e) return;
    int s = src[ed], d = dst[ed];
    float w = dinv[s] * dinv[d];
    float4 v = ((const float4*)(h + (size_t)s * IN_CH))[lane];
    float* o = out + (size_t)d * IN_CH + lane * 4;
    atomicAdd(o + 0, v.x * w);
    atomicAdd(o + 1, v.y * w);
    atomicAdd(o + 2, v.z * w);
    atomicAdd(o + 3, v.w * w);
}

// ---------------- z = h @ W^T + bias via V_WMMA_F32_16X16X4_F32 ----------------
// one wave computes a 16x16 tile of z; K=128 -> 32 wmma steps
// requires N % 16 == 0 (50000 = 3125*16), OUT_CH = 128 = 8*16

__global__ void gemm_z_kernel(const float* __restrict__ h,
                              const float* __restrict__ W,     // [OUT_CH, IN_CH] row-major
                              const float* __restrict__ bias,  // [OUT_CH]
                              float* __restrict__ z, int nTiles) {
    int wave = (blockIdx.x * blockDim.x + threadIdx.x) >> 5;  // wave-uniform
    int lane = threadIdx.x & 31;
    if (wave >= nTiles) return;                               // uniform exit, EXEC stays full
    int mt = wave >> 3;          // row tile
    int nt = wave & 7;           // col tile
    int half = lane >> 4;        // 0: K+{0,1}, 1: K+{2,3}
    int mrow = lane & 15;

    const float* arow = h + (size_t)(mt * 16 + mrow) * IN_CH + half * 2;
    const float* brow = W + (size_t)(nt * 16 + mrow) * IN_CH + half * 2; // B[k][col]=W[col][k]

    v8f acc = {};
    for (int kk = 0; kk < 32; ++kk) {
        int k = kk * 4;
        v2f a; a.x = arow[k]; a.y = arow[k + 1];
        v2f b; b.x = brow[k]; b.y = brow[k + 1];
        acc = wmma4_f32(a, b, acc);
    }
    // D layout: VGPR v, lane -> row (v + 8*half), col mrow within tile
    float bcol = bias[nt * 16 + mrow];
    float* zp = z + (size_t)(mt * 16 + 8 * half) * OUT_CH + nt * 16 + mrow;
#pragma unroll
    for (int v = 0; v < 8; ++v) zp[v * OUT_CH] = acc[v] + bcol;
}

// ---------------- decode: r = relu((z[a]*z[b]) @ w1^T + b1) @ w2^T + b2 -------
// one wave handles 16 label edges: 16x128 A (on-the-fly product) times 128x64,
// 4 accumulator tiles, then fused relu + w2 dot + cross-lane reduction

__global__ void decode_kernel(const float* __restrict__ z,
                              const int* __restrict__ eli,   // [2, L] flat
                              const float* __restrict__ w1,  // [64, 128]
                              const float* __restrict__ bias1, // [64]
                              const float* __restrict__ wout,  // [64]
                              const float* __restrict__ bias2, // [1]
                              float* __restrict__ r, int nTiles, int L) {
    int wave = (blockIdx.x * blockDim.x + threadIdx.x) >> 5;  // wave-uniform
    int lane = threadIdx.x & 31;
    if (wave >= nTiles) return;                               // uniform exit
    int half = lane >> 4;
    int mrow = lane & 15;

    int e = wave * 16 + mrow;
    int ai = eli[e];
    int bi = eli[L + e];
    const float* za = z + (size_t)ai * OUT_CH + half * 2;
    const float* zb = z + (size_t)bi * OUT_CH + half * 2;
    const float* w1p = w1 + (size_t)mrow * IN_CH + half * 2;  // col = t*16 + mrow

    v8f acc[4] = {};
    for (int kk = 0; kk < 32; ++kk) {
        int k = kk * 4;
        v2f a; a.x = za[k] * zb[k]; a.y = za[k + 1] * zb[k + 1];
#pragma unroll
        for (int t = 0; t < 4; ++t) {
            v2f b; b.x = w1p[t * 16 * IN_CH + k]; b.y = w1p[t * 16 * IN_CH + k + 1];
            acc[t] = wmma4_f32(a, b, acc[t]);
        }
    }

    // per-lane partials: lane holds rows (v+8*half), cols {t*16+mrow}
    float p[8];
#pragma unroll
    for (int v = 0; v < 8; ++v) p[v] = 0.0f;
#pragma unroll
    for (int t = 0; t < 4; ++t) {
        int col = t * 16 + mrow;
        float wj = wout[col];
        float bj = bias1[col];
#pragma unroll
        for (int v = 0; v < 8; ++v) {
            float hv = acc[t][v] + bj;
            p[v] += fmaxf(hv, 0.0f) * wj;
        }
    }
    // reduce the 16 columns across each 16-lane half
#pragma unroll
    for (int v = 0; v < 8; ++v) {
        float s = p[v];
        s += __shfl_xor(s, 1, 16);
        s += __shfl_xor(s, 2, 16);
        s += __shfl_xor(s, 4, 16);
        s += __shfl_xor(s, 8, 16);
        p[v] = s;
    }
    if (mrow == 0) {
        float b2v = bias2[0];
#pragma unroll
        for (int v = 0; v < 8; ++v)
            r[wave * 16 + v + 8 * half] = p[v] + b2v;
    }
}

extern "C" void kernel_launch(void* const* d_in, const int* in_sizes, int n_in,
                              void* d_out, int out_size, void* d_ws, size_t ws_size,
                              hipStream_t stream) {
    const float* x   = (const float*)d_in[0];
    const int*   ei  = (const int*)d_in[1];
    const int*   eli = (const int*)d_in[2];
    const float* Wc  = (const float*)d_in[3];
    const float* bc  = (const float*)d_in[4];
    const float* w1  = (const float*)d_in[5];
    const float* b1  = (const float*)d_in[6];
    const float* w2  = (const float*)d_in[7];
    const float* b2  = (const float*)d_in[8];
    float* r = (float*)d_out;

    const int N = in_sizes[0] / IN_CH;   // 50000
    const int E = in_sizes[1] / 2;       // 600000
    const int L = in_sizes[2] / 2;       // 200000

    float* dinv = (float*)d_ws;                      // N floats (holds deg then dinv)
    float* bufA = dinv + N;                          // N*128 (offset 16B-aligned: N*4 % 16 == 0)
    float* bufB = bufA + (size_t)N * IN_CH;          // N*128
    const int* src = ei;
    const int* dst = ei + E;

    const int TB = 256;
    // degree + rsqrt
    deg_init_kernel<<<(N + TB - 1) / TB, TB, 0, stream>>>(dinv, N);
    deg_accum_kernel<<<(E + TB - 1) / TB, TB, 0, stream>>>(dst, dinv, E);
    deg_rsqrt_kernel<<<(N + TB - 1) / TB, TB, 0, stream>>>(dinv, N);

    // hop 1: x -> bufA
    {
        long long t = (long long)N * 32;
        hop_init_kernel<<<(unsigned)((t + TB - 1) / TB), TB, 0, stream>>>(x, dinv, bufA, N);
        long long te = (long long)E * 32;
        hop_scatter_kernel<<<(unsigned)((te + TB - 1) / TB), TB, 0, stream>>>(x, dinv, src, dst, bufA, E);
    }
    // hop 2: bufA -> bufB
    {
        long long t = (long long)N * 32;
        hop_init_kernel<<<(unsigned)((t + TB - 1) / TB), TB, 0, stream>>>(bufA, dinv, bufB, N);
        long long te = (long long)E * 32;
        hop_scatter_kernel<<<(unsigned)((te + TB - 1) / TB), TB, 0, stream>>>(bufA, dinv, src, dst, bufB, E);
    }

    // z = bufB @ Wc^T + bc  -> bufA   (N%16==0, OUT_CH==128)
    {
        int zTiles = (N / 16) * (OUT_CH / 16);       // 25000
        long long t = (long long)zTiles * 32;
        gemm_z_kernel<<<(unsigned)((t + TB - 1) / TB), TB, 0, stream>>>(bufB, Wc, bc, bufA, zTiles);
    }

    // decode -> d_out
    {
        int dTiles = L / 16;                          // 12500
        long long t = (long long)dTiles * 32;
        decode_kernel<<<(unsigned)((t + TB - 1) / TB), TB, 0, stream>>>(bufA, eli, w1, b1, w2, b2, r, dTiles, L);
    }
}